// WordExtractionBidirectional_79448305042054
// MI455X (gfx1250) — compile-verified
//
#include <hip/hip_runtime.h>

#ifndef __has_builtin
#define __has_builtin(x) 0
#endif

// Problem constants from the reference: B=16, T=4096, D=256, W=512
#define B_ 16
#define T_ 4096
#define D_ 256
#define W_ 512

typedef float float4v __attribute__((ext_vector_type(4)));
typedef int   int4v   __attribute__((ext_vector_type(4)));
// Exact parameter types per hipcc diagnostic: generic int4* (global side),
// and the LDS side must be an addrspace(3) typed pointer.
typedef __attribute__((address_space(3))) int4v lds_int4;

#if __has_builtin(__builtin_amdgcn_global_load_async_to_lds_b128) && \
    __has_builtin(__builtin_amdgcn_global_store_async_from_lds_b128) && \
    __has_builtin(__builtin_amdgcn_s_wait_asynccnt)
#define USE_ASYNC_LDS 1
#else
#define USE_ASYNC_LDS 0
#endif

// One 128-thread block per output row (b,w).
//   threads   0..63  : fwd half, float4 j = tid       -> out[..][0:256]
//   threads  64..127 : bwd half, float4 j = tid - 64  -> out[..][256:512]
// The fwd/bwd split is wave-uniform under wave32 (waves 0-1 fwd, waves 2-3 bwd).
// Indices are block-uniform -> compiler emits scalar s_load for them.
__global__ __launch_bounds__(128) void gather_concat_b128_kernel(
    const float* __restrict__ x_fwd, const float* __restrict__ x_bwd,
    const int* __restrict__ idx_fwd, const int* __restrict__ idx_back,
    float* __restrict__ out)
{
  const int tid = threadIdx.x;
  const int bw  = blockIdx.x;            // 0 .. B*W-1, row id
  const int b   = bw >> 9;               // bw / W_

  // Block-uniform index loads + masking (idx < 2 -> gather timestep 0).
  int i_f = idx_fwd[bw];
  int i_b = idx_back[bw];
  i_f = (i_f >= 2) ? i_f : 0;
  i_b = (i_b >= 2) ? i_b : 0;

  const bool is_bwd = (tid >= 64);       // wave-uniform
  const int  j      = tid & 63;          // float4 slot within the 256-float row

  const float* src = is_bwd
      ? (x_bwd + ((size_t)b * T_ + (size_t)i_b) * D_)
      : (x_fwd + ((size_t)b * T_ + (size_t)i_f) * D_);
  float* dst = out + (size_t)bw * (2 * D_) + (is_bwd ? D_ : 0);

#if USE_ASYNC_LDS
  // CDNA5 async DMA path: Global -> LDS -> Global, no VGPR round-trip.
  // Each lane moves one float4; a wave issues one B128 async op = 512 B.
  __shared__ __align__(16) float lds_buf[128 * 4];   // 2 KB per block

  __builtin_amdgcn_global_load_async_to_lds_b128(
      (int4v*)(src + (size_t)j * 4),
      (lds_int4*)&lds_buf[tid * 4],
      /*offset=*/0, /*cpol=*/0);
  // Wait for this wave's async load to land in LDS before the store reads it.
  __builtin_amdgcn_s_wait_asynccnt(0);
  __builtin_amdgcn_global_store_async_from_lds_b128(
      (int4v*)(dst + (size_t)j * 4),
      (lds_int4*)&lds_buf[tid * 4],
      /*offset=*/0, /*cpol=*/0);
  __builtin_amdgcn_s_wait_asynccnt(0);
#else
  // Fallback: direct fully-coalesced B128 load/store through VGPRs.
  float4v v = *(const float4v*)(src + (size_t)j * 4);
  *(float4v*)(dst + (size_t)j * 4) = v;
#endif
}

extern "C" void kernel_launch(void* const* d_in, const int* in_sizes, int n_in,
                              void* d_out, int out_size, void* d_ws, size_t ws_size,
                              hipStream_t stream) {
  (void)in_sizes; (void)n_in; (void)out_size; (void)d_ws; (void)ws_size;
  const float* x_fwd    = (const float*)d_in[0];
  const float* x_bwd    = (const float*)d_in[1];
  const int*   idx_fwd  = (const int*)d_in[2];
  const int*   idx_back = (const int*)d_in[3];
  float*       out      = (float*)d_out;

  dim3 grid(B_ * W_);   // 8192 blocks: one per output row
  dim3 block(128);      // 4 wave32 waves; every output element written exactly once
  hipLaunchKernelGGL(gather_concat_b128_kernel, grid, block, 0, stream,
                     x_fwd, x_bwd, idx_fwd, idx_back, out);
}